// CustomGAT_36240934043837
// MI455X (gfx1250) — compile-verified
//
#include <hip/hip_runtime.h>
#include <math.h>

typedef __attribute__((ext_vector_type(2))) float v2f;
typedef __attribute__((ext_vector_type(8))) float v8f;

#define NEG_SLOPE 0.2f

// Monotone float -> uint key so atomicMax(u32) implements float max.
__device__ __forceinline__ unsigned fkey(float f) {
    unsigned b = __float_as_uint(f);
    return (b & 0x80000000u) ? ~b : (b | 0x80000000u);
}
__device__ __forceinline__ float fkey_inv(unsigned u) {
    return (u & 0x80000000u) ? __uint_as_float(u & 0x7fffffffu)
                             : __uint_as_float(~u);
}
// fkey(-inf) == 0x007fffff
#define FKEY_NEG_INF 0x007fffffu

// ---------------------------------------------------------------------------
// fp32 GEMM via V_WMMA_F32_16X16X4_F32.  One wave computes a 16-row slab of
// C = A[N,K] x B[K,NCOL], covering ALL NCOL/16 column tiles so the A fragment
// (the only HBM-resident operand) is loaded exactly once.
// A-frag (16x4 f32): lanes 0-15 row M=l hold K=0,1; lanes 16-31 hold K=2,3.
// B-frag (4x16 f32): lanes 0-15 col N=l hold K=0,1; lanes 16-31 hold K=2,3.
// C (16x16 f32): VGPR i = row i (lanes 0-15) / row i+8 (lanes 16-31), col = l.
// N must be a multiple of 16 (100000 = 6250*16).
// ---------------------------------------------------------------------------
template <int K, int NCOL>
__global__ void gemm_wmma_kernel(const float* __restrict__ A,
                                 const float* __restrict__ B,
                                 float* __restrict__ C, int N) {
    constexpr int NT = NCOL / 16;          // column tiles per wave
    const int lane = threadIdx.x & 31;
    const int l    = lane & 15;
    const int hf   = lane >> 4;            // 0: K+0/1, 1: K+2/3
    const int row0 = blockIdx.x * 16;
    if (row0 >= N) return;

    const float* __restrict__ arow = A + (size_t)(row0 + l) * K;
    v8f acc[NT];
#pragma unroll
    for (int t = 0; t < NT; ++t) acc[t] = (v8f){};

#pragma unroll
    for (int k = 0; k < K; k += 4) {
        v2f a;
        a.x = arow[k + 2 * hf];
        a.y = arow[k + 2 * hf + 1];
#pragma unroll
        for (int t = 0; t < NT; ++t) {
            v2f b;
            b.x = B[(size_t)(k + 2 * hf) * NCOL + 16 * t + l];
            b.y = B[(size_t)(k + 2 * hf + 1) * NCOL + 16 * t + l];
            acc[t] = __builtin_amdgcn_wmma_f32_16x16x4_f32(
                /*neg_a=*/false, a, /*neg_b=*/false, b,
                /*c_mod=*/(short)0, acc[t], /*reuse_a=*/false,
                /*reuse_b=*/false);
        }
    }
#pragma unroll
    for (int t = 0; t < NT; ++t)
#pragma unroll
        for (int i = 0; i < 8; ++i)
            C[(size_t)(row0 + i + 8 * hf) * NCOL + 16 * t + l] = acc[t][i];
}

// alpha_s[n,h] = sum_c xw[n,h,c]*a_src[h,c];  alpha_d likewise.
template <int H, int C>
__global__ void alpha_kernel(const float* __restrict__ xw,
                             const float* __restrict__ a_src,
                             const float* __restrict__ a_dst,
                             float* __restrict__ as_, float* __restrict__ ad_,
                             int N) {
    int idx = blockIdx.x * blockDim.x + threadIdx.x;
    if (idx >= N * H) return;
    int n = idx / H, h = idx % H;
    const float* v = xw + (size_t)n * (H * C) + h * C;
    float s = 0.f, d = 0.f;
#pragma unroll
    for (int c = 0; c < C; ++c) {
        s += v[c] * a_src[h * C + c];
        d += v[c] * a_dst[h * C + c];
    }
    as_[idx] = s;
    ad_[idx] = d;
}

// Zero accumulators / denominators, init max keys to -inf.
__global__ void init_kernel(unsigned* __restrict__ m_, float* __restrict__ den_,
                            float* __restrict__ acc_, int NH, int NHC) {
    int idx = blockIdx.x * blockDim.x + threadIdx.x;
    if (idx < NHC) acc_[idx] = 0.f;
    if (idx < NH) { m_[idx] = FKEY_NEG_INF; den_[idx] = 0.f; }
}

__device__ __forceinline__ void edge_sd(const int* __restrict__ srcs,
                                        const int* __restrict__ dsts, int E,
                                        int i, int& s, int& d) {
    if (i < E) { s = srcs[i]; d = dsts[i]; }
    else       { s = i - E;   d = i - E;   }   // appended self-loop
}

template <int H>
__global__ void edge_max_kernel(const int* __restrict__ srcs,
                                const int* __restrict__ dsts, int E, int N,
                                const float* __restrict__ as_,
                                const float* __restrict__ ad_,
                                unsigned* __restrict__ m_) {
    int i = blockIdx.x * blockDim.x + threadIdx.x;
    if (i >= E + N) return;
    int s, d; edge_sd(srcs, dsts, E, i, s, d);
#pragma unroll
    for (int h = 0; h < H; ++h) {
        float e = as_[s * H + h] + ad_[d * H + h];
        e = (e > 0.f) ? e : NEG_SLOPE * e;
        atomicMax(&m_[d * H + h], fkey(e));
    }
}

template <int H>
__global__ void edge_denom_kernel(const int* __restrict__ srcs,
                                  const int* __restrict__ dsts, int E, int N,
                                  const float* __restrict__ as_,
                                  const float* __restrict__ ad_,
                                  const unsigned* __restrict__ m_,
                                  float* __restrict__ den_) {
    int i = blockIdx.x * blockDim.x + threadIdx.x;
    if (i >= E + N) return;
    int s, d; edge_sd(srcs, dsts, E, i, s, d);
#pragma unroll
    for (int h = 0; h < H; ++h) {
        float e = as_[s * H + h] + ad_[d * H + h];
        e = (e > 0.f) ? e : NEG_SLOPE * e;
        atomicAdd(&den_[d * H + h], expf(e - fkey_inv(m_[d * H + h])));
    }
}

template <int H, int C>
__global__ void edge_accum_kernel(const int* __restrict__ srcs,
                                  const int* __restrict__ dsts, int E, int N,
                                  const float* __restrict__ as_,
                                  const float* __restrict__ ad_,
                                  const unsigned* __restrict__ m_,
                                  const float* __restrict__ den_,
                                  const float* __restrict__ xw,
                                  float* __restrict__ acc_) {
    static_assert(C % 4 == 0, "C must be a multiple of 4");
    int i = blockIdx.x * blockDim.x + threadIdx.x;
    if (i >= E + N) return;
    int s, d; edge_sd(srcs, dsts, E, i, s, d);

    float w[H];
#pragma unroll
    for (int h = 0; h < H; ++h) {
        float e = as_[s * H + h] + ad_[d * H + h];
        e = (e > 0.f) ? e : NEG_SLOPE * e;
        w[h] = expf(e - fkey_inv(m_[d * H + h])) / den_[d * H + h];
    }
    // Gather the full H*C feature row with 128-bit loads, scatter with atomics.
    const float4* __restrict__ sv =
        (const float4*)(xw + (size_t)s * (H * C));
    float* __restrict__ dv = acc_ + (size_t)d * (H * C);
#pragma unroll
    for (int f = 0; f < (H * C) / 4; ++f) {
        float4 v = sv[f];
        float wh = w[(f * 4) / C];            // C % 4 == 0: one head per vec
        atomicAdd(&dv[f * 4 + 0], wh * v.x);
        atomicAdd(&dv[f * 4 + 1], wh * v.y);
        atomicAdd(&dv[f * 4 + 2], wh * v.z);
        atomicAdd(&dv[f * 4 + 3], wh * v.w);
    }
}

__global__ void bias_elu_kernel(const float* __restrict__ acc_,
                                const float* __restrict__ bias,
                                float* __restrict__ h_, int total, int F) {
    int i = blockIdx.x * blockDim.x + threadIdx.x;
    if (i >= total) return;
    float v = acc_[i] + bias[i & (F - 1)];    // F is a power of two (32)
    h_[i] = (v > 0.f) ? v : (expf(v) - 1.f);
}

__global__ void logsoftmax_kernel(const float* __restrict__ acc_,
                                  const float* __restrict__ bias,
                                  float* __restrict__ out, int N) {
    int n = blockIdx.x * blockDim.x + threadIdx.x;
    if (n >= N) return;
    float v[16];
    float m = -INFINITY;
#pragma unroll
    for (int c = 0; c < 16; ++c) {
        v[c] = acc_[(size_t)n * 16 + c] + bias[c];
        m = fmaxf(m, v[c]);
    }
    float ssum = 0.f;
#pragma unroll
    for (int c = 0; c < 16; ++c) ssum += expf(v[c] - m);
    float lg = m + logf(ssum);
#pragma unroll
    for (int c = 0; c < 16; ++c) out[(size_t)n * 16 + c] = v[c] - lg;
}

extern "C" void kernel_launch(void* const* d_in, const int* in_sizes, int n_in,
                              void* d_out, int out_size, void* d_ws,
                              size_t ws_size, hipStream_t stream) {
    const float* x   = (const float*)d_in[0];
    const int*   ei  = (const int*)d_in[1];
    const float* W1  = (const float*)d_in[2];
    const float* a1s = (const float*)d_in[3];
    const float* a1d = (const float*)d_in[4];
    const float* b1  = (const float*)d_in[5];
    const float* W2  = (const float*)d_in[6];
    const float* a2s = (const float*)d_in[7];
    const float* a2d = (const float*)d_in[8];
    const float* b2  = (const float*)d_in[9];
    float* out = (float*)d_out;

    const int N = in_sizes[0] / 128;   // 100000
    const int E = in_sizes[1] / 2;     // 1600000
    const int* srcs = ei;
    const int* dsts = ei + E;
    const int T = E + N;               // with self-loops

    // Carve workspace (256B aligned chunks).
    char* p = (char*)d_ws;
    auto carve = [&](size_t bytes) {
        void* r = (void*)p;
        p += (bytes + 255) & ~(size_t)255;
        return r;
    };
    float*    xw1  = (float*)carve((size_t)N * 32 * 4);
    float*    as1  = (float*)carve((size_t)N * 4 * 4);
    float*    ad1  = (float*)carve((size_t)N * 4 * 4);
    unsigned* m1   = (unsigned*)carve((size_t)N * 4 * 4);
    float*    den1 = (float*)carve((size_t)N * 4 * 4);
    float*    out1 = (float*)carve((size_t)N * 32 * 4);
    float*    h1   = (float*)carve((size_t)N * 32 * 4);
    float*    xw2  = (float*)carve((size_t)N * 16 * 4);
    float*    as2  = (float*)carve((size_t)N * 4);
    float*    ad2  = (float*)carve((size_t)N * 4);
    unsigned* m2   = (unsigned*)carve((size_t)N * 4);
    float*    den2 = (float*)carve((size_t)N * 4);
    float*    out2 = (float*)carve((size_t)N * 16 * 4);

    const int B = 256;
    auto g = [](int n, int b) { return (n + b - 1) / b; };

    // ---- Layer 1: xw1 = x @ W1 (WMMA fp32, A read once), H=4, C=8 ----
    gemm_wmma_kernel<128, 32><<<N / 16, 32, 0, stream>>>(x, W1, xw1, N);
    alpha_kernel<4, 8><<<g(N * 4, B), B, 0, stream>>>(xw1, a1s, a1d, as1, ad1, N);
    init_kernel<<<g(N * 32, B), B, 0, stream>>>(m1, den1, out1, N * 4, N * 32);
    edge_max_kernel<4><<<g(T, B), B, 0, stream>>>(srcs, dsts, E, N, as1, ad1, m1);
    edge_denom_kernel<4><<<g(T, B), B, 0, stream>>>(srcs, dsts, E, N, as1, ad1,
                                                    m1, den1);
    edge_accum_kernel<4, 8><<<g(T, B), B, 0, stream>>>(srcs, dsts, E, N, as1,
                                                       ad1, m1, den1, xw1, out1);
    bias_elu_kernel<<<g(N * 32, B), B, 0, stream>>>(out1, b1, h1, N * 32, 32);

    // ---- Layer 2: xw2 = h1 @ W2 (WMMA fp32), H=1, C=16 ----
    gemm_wmma_kernel<32, 16><<<N / 16, 32, 0, stream>>>(h1, W2, xw2, N);
    alpha_kernel<1, 16><<<g(N, B), B, 0, stream>>>(xw2, a2s, a2d, as2, ad2, N);
    init_kernel<<<g(N * 16, B), B, 0, stream>>>(m2, den2, out2, N, N * 16);
    edge_max_kernel<1><<<g(T, B), B, 0, stream>>>(srcs, dsts, E, N, as2, ad2, m2);
    edge_denom_kernel<1><<<g(T, B), B, 0, stream>>>(srcs, dsts, E, N, as2, ad2,
                                                    m2, den2);
    edge_accum_kernel<1, 16><<<g(T, B), B, 0, stream>>>(srcs, dsts, E, N, as2,
                                                        ad2, m2, den2, xw2, out2);
    logsoftmax_kernel<<<g(N, B), B, 0, stream>>>(out2, b2, out, N);
}